// GradientLayer_4904852652333
// MI455X (gfx1250) — compile-verified
//
#include <hip/hip_runtime.h>

// PINN jet-propagation kernel for gfx1250 (MI455X).
// Forward Taylor-mode AD: 10 derivative channels per point pushed through the
// MLP. Hidden-layer matmuls run on V_WMMA_F32_16X16X4_F32 (full fp32 matrix
// pipe -- precision needed for 3rd derivatives). Activations + weights staged
// in LDS (WGP has 320 KB).

typedef float v2f __attribute__((ext_vector_type(2)));
typedef float v8f __attribute__((ext_vector_type(8)));

#define WIDTH    128
#define NCH      10          // jet channels: v,x,y,xx,xy,yy,xxx,xxy,xyy,yyy
#define PTS      16          // points per workgroup
#define NROWS    (NCH*PTS)   // 160 rows
#define HS       132         // padded LDS row stride (floats) -> conflict-free b64
#define WP       144         // padded pair-row pitch of W in float2 units
#define NTHREADS 320         // 10 waves; wave w owns channel row-tile w

__device__ __forceinline__ void tanh_jet(float a, float ax, float ay,
                                         float axx, float axy, float ayy,
                                         float axxx, float axxy, float axyy, float ayyy,
                                         float* o)
{
    float t  = tanhf(a);
    float s  = 1.0f - t * t;           // t'
    float d2 = -2.0f * t * s;          // t''
    float d3 = -2.0f * (s * s + t * d2); // t'''
    o[0] = t;
    o[1] = s * ax;
    o[2] = s * ay;
    o[3] = d2 * ax * ax + s * axx;
    o[4] = d2 * ax * ay + s * axy;
    o[5] = d2 * ay * ay + s * ayy;
    o[6] = d3 * ax * ax * ax + 3.0f * d2 * ax * axx + s * axxx;
    o[7] = d3 * ax * ax * ay + d2 * (2.0f * ax * axy + ay * axx) + s * axxy;
    o[8] = d3 * ax * ay * ay + d2 * (2.0f * ay * axy + ax * ayy) + s * axyy;
    o[9] = d3 * ay * ay * ay + 3.0f * d2 * ay * ayy + s * ayyy;
}

__global__ __launch_bounds__(NTHREADS)
void pinn_jet_wmma(const float* __restrict__ xy,
                   const float* __restrict__ W0, const float* __restrict__ b0,
                   const float* __restrict__ W1, const float* __restrict__ b1,
                   const float* __restrict__ W2, const float* __restrict__ b2,
                   const float* __restrict__ W3, const float* __restrict__ b3,
                   const float* __restrict__ W4, const float* __restrict__ b4,
                   const float* __restrict__ W5, const float* __restrict__ b5,
                   const float* __restrict__ W6, const float* __restrict__ b6,
                   float* __restrict__ out, int npts)
{
    __shared__ __align__(16) float Hbuf[2][NROWS * HS];   // 165 KB: jet activations, double-buffered
    __shared__ v2f   Wl[32 * 2 * WP];                     // 72 KB: layer weights in B-fragment pair layout
    __shared__ float W6s[WIDTH * 2];
    __shared__ float Sout[PTS * NCH * 2];

    const int tid  = threadIdx.x;
    const int wave = tid >> 5;          // 0..9 == channel row-tile
    const int lane = tid & 31;
    const int l16  = lane & 15;
    const int half = lane >> 4;         // K-half within a 4-wide K block
    const int p0   = blockIdx.x * PTS;

    const float* Whid[5] = {W1, W2, W3, W4, W5};
    const float* bhid[5] = {b1, b2, b3, b4, b5};

    // ---------------- Layer 0: 2 -> 128, analytic input jet ----------------
    for (int idx = tid; idx < PTS * WIDTH; idx += NTHREADS) {
        int pt = idx >> 7, u = idx & 127;
        int gp = p0 + pt;
        float x = 0.0f, y = 0.0f;
        if (gp < npts) { x = xy[gp * 2 + 0]; y = xy[gp * 2 + 1]; }
        float wx = W0[u], wy = W0[WIDTH + u];
        float a  = fmaf(x, wx, fmaf(y, wy, b0[u]));
        float o[10];
        // input jet: a_x=wx, a_y=wy, all higher pre-activation derivs zero
        tanh_jet(a, wx, wy, 0.0f, 0.0f, 0.0f, 0.0f, 0.0f, 0.0f, 0.0f, o);
        float* H = Hbuf[0];
        #pragma unroll
        for (int c = 0; c < NCH; ++c) H[(c * PTS + pt) * HS + u] = o[c];
    }
    __syncthreads();

    // ---------------- Hidden layers 1..5: 128 -> 128 via fp32 WMMA ----------------
    for (int l = 1; l <= 5; ++l) {
        const float* Wg = Whid[l - 1];
        // Stage W[k][n] into B-fragment pair layout:
        // Wl[(kb*2+h)*WP + n] = (W[4kb+2h][n], W[4kb+2h+1][n])
        for (int p = tid; p < 32 * 2 * WIDTH; p += NTHREADS) {
            int kb = p >> 8;
            int h  = (p >> 7) & 1;
            int n  = p & 127;
            int k  = 4 * kb + 2 * h;
            v2f w;
            w.x = Wg[k * WIDTH + n];
            w.y = Wg[(k + 1) * WIDTH + n];
            Wl[(kb * 2 + h) * WP + n] = w;
        }
        const float* Hin  = Hbuf[(l - 1) & 1];
        float*       Hout = Hbuf[l & 1];
        __syncthreads();

        v8f acc[8] = {};   // 8 N-tiles of 16 -> full 128-wide output row block

        // A fragment: lane -> row (wave*16 + l16), K pair (4kb + 2*half)
        const float* arow = &Hin[(wave * PTS + l16) * HS + 2 * half];
        #pragma unroll 4
        for (int kb = 0; kb < 32; ++kb) {
            v2f afrag = *(const v2f*)(arow + 4 * kb);
            const v2f* brow = &Wl[(kb * 2 + half) * WP + l16];
            #pragma unroll
            for (int n = 0; n < 8; ++n) {
                v2f bfrag = brow[n * 16];
                acc[n] = __builtin_amdgcn_wmma_f32_16x16x4_f32(
                    /*neg_a=*/false, afrag, /*neg_b=*/false, bfrag,
                    /*c_mod=*/(short)0, acc[n], /*reuse_a=*/false, /*reuse_b=*/false);
            }
        }

        // D layout: VGPR i holds D[M = i + 8*half][N = l16] -> scatter to LDS
        #pragma unroll
        for (int n = 0; n < 8; ++n) {
            float* dst = &Hout[(wave * PTS + 8 * half) * HS + n * 16 + l16];
            #pragma unroll
            for (int i = 0; i < 8; ++i) dst[i * HS] = acc[n][i];
        }
        __syncthreads();

        // Elementwise tanh-jet composition, in place on Hout (bias -> value channel only)
        const float* bl = bhid[l - 1];
        for (int idx = tid; idx < PTS * WIDTH; idx += NTHREADS) {
            int pt = idx >> 7, u = idx & 127;
            float a    = Hout[(0 * PTS + pt) * HS + u] + bl[u];
            float ax   = Hout[(1 * PTS + pt) * HS + u];
            float ay   = Hout[(2 * PTS + pt) * HS + u];
            float axx  = Hout[(3 * PTS + pt) * HS + u];
            float axy  = Hout[(4 * PTS + pt) * HS + u];
            float ayy  = Hout[(5 * PTS + pt) * HS + u];
            float axxx = Hout[(6 * PTS + pt) * HS + u];
            float axxy = Hout[(7 * PTS + pt) * HS + u];
            float axyy = Hout[(8 * PTS + pt) * HS + u];
            float ayyy = Hout[(9 * PTS + pt) * HS + u];
            float o[10];
            tanh_jet(a, ax, ay, axx, axy, ayy, axxx, axxy, axyy, ayyy, o);
            #pragma unroll
            for (int c = 0; c < NCH; ++c) Hout[(c * PTS + pt) * HS + u] = o[c];
        }
        __syncthreads();
    }

    // ---------------- Head: 128 -> 2, one dot product per thread ----------------
    for (int p = tid; p < WIDTH * 2; p += NTHREADS) W6s[p] = W6[p];
    __syncthreads();

    const float* Hlast = Hbuf[1];   // layer 5 output lives in buffer 1
    {
        int pt  = tid / 20;         // 320 threads = 16 pts x 10 ch x 2 outs
        int rem = tid % 20;
        int ch  = rem >> 1, o = rem & 1;
        float dot = (ch == 0) ? b6[o] : 0.0f;   // bias only on the value channel
        const float* hr = &Hlast[(ch * PTS + pt) * HS];
        #pragma unroll 8
        for (int k = 0; k < WIDTH; ++k) dot = fmaf(hr[k], W6s[k * 2 + o], dot);
        Sout[pt * 20 + rem] = dot;
    }
    __syncthreads();

    // Assemble the 14 reference outputs:
    // [psi, p, p_x, p_y, u, u_x, u_y, u_xx, u_yy, v, v_x, v_y, v_xx, v_yy]
    if (tid < PTS * 14) {
        int pt = tid / 14, j = tid % 14;
        const int   chT[14] = {0, 0, 1, 2, 2, 4, 5, 7, 9, 1, 3, 4, 6, 8};
        const int   oT[14]  = {0, 1, 1, 1, 0, 0, 0, 0, 0, 0, 0, 0, 0, 0};
        const float sgT[14] = {1, 1, 1, 1, 1, 1, 1, 1, 1, -1, -1, -1, -1, -1};
        int gp = p0 + pt;
        if (gp < npts)
            out[gp * 14 + j] = sgT[j] * Sout[pt * 20 + chT[j] * 2 + oT[j]];
    }
}

extern "C" void kernel_launch(void* const* d_in, const int* in_sizes, int n_in,
                              void* d_out, int out_size, void* d_ws, size_t ws_size,
                              hipStream_t stream)
{
    const float* xy = (const float*)d_in[0];
    int npts = in_sizes[0] / 2;
    const float* W[7];
    const float* B[7];
    for (int i = 0; i < 7; ++i) {
        W[i] = (const float*)d_in[1 + 2 * i];
        B[i] = (const float*)d_in[2 + 2 * i];
    }
    int blocks = (npts + PTS - 1) / PTS;
    pinn_jet_wmma<<<blocks, NTHREADS, 0, stream>>>(
        xy, W[0], B[0], W[1], B[1], W[2], B[2], W[3], B[3],
        W[4], B[4], W[5], B[5], W[6], B[6], (float*)d_out, npts);
}